// CrossCBR_41369124995110
// MI455X (gfx1250) — compile-verified
//
#include <hip/hip_runtime.h>
#include <math.h>
#include <stdint.h>

// CrossCBR propagation for MI455X (gfx1250, wave32).
// Memory-bound SpMM scatter/gather pipeline (arith intensity ~0.25 flop/B vs
// 23.3 TB/s HBM -> pure bandwidth; WMMA has no dense tile structure to use).
// CDNA5-specific paths used:
//   * global_load_async_to_lds_b32 + s_wait_asynccnt: DMA edge-index chunks
//     into LDS (kills the 64x-redundant per-lane index loads)
//   * ds_load of staged indices, wave32 __shfl_xor row-norm reduction
//   * global_prefetch_b8 lookahead on the randomly gathered feature rows
//   * b128 vector loads/stores for all dense passes
//   * f32 global atomics for the scatter (RMW resolved in the 192MB L2,
//     which fully holds the 76MB accumulation buffer)

#define D64 64
#define NU_C 100000
#define NI_C 200000
#define NB_C 50000
#define CHUNK 256   // edges staged per block (2KB LDS)

__global__ __launch_bounds__(256)
void fill_zero4_k(float4* __restrict__ p, long n4) {
    long i = (long)blockIdx.x * blockDim.x + threadIdx.x;
    if (i < n4) p[i] = make_float4(0.f, 0.f, 0.f, 0.f);
}

// dst[r*stride4 + k] = src[r*16 + k]  (float4 granularity; dst pre-offset)
__global__ __launch_bounds__(256)
void copy_rows4_k(float4* __restrict__ dst, int dstStride4,
                  const float4* __restrict__ src, int nRows) {
    int idx = blockIdx.x * blockDim.x + threadIdx.x;
    if (idx >= nRows * 16) return;
    int r = idx >> 4, k = idx & 15;
    dst[(size_t)r * dstStride4 + k] = src[idx];
}

__global__ __launch_bounds__(256)
void zero_rows4_k(float4* __restrict__ dst, int dstStride4, int nRows) {
    int idx = blockIdx.x * blockDim.x + threadIdx.x;
    if (idx >= nRows * 16) return;
    int r = idx >> 4, k = idx & 15;
    dst[(size_t)r * dstStride4 + k] = make_float4(0.f, 0.f, 0.f, 0.f);
}

__global__ __launch_bounds__(256)
void scale_rows4_k(float4* __restrict__ dst, int dstStride4,
                   const float* __restrict__ bsize, int nRows) {
    int idx = blockIdx.x * blockDim.x + threadIdx.x;
    if (idx >= nRows * 16) return;
    int r = idx >> 4, k = idx & 15;
    float s = 1.0f / (bsize[r] + 1e-8f);
    float4 v = dst[(size_t)r * dstStride4 + k];
    v.x *= s; v.y *= s; v.z *= s; v.w *= s;
    dst[(size_t)r * dstStride4 + k] = v;
}

// Stage one CHUNK of edge indices into LDS via the gfx1250 async DMA path.
// Each enabled lane issues one async b32 per array; data lands in LDS without
// touching VGPRs; s_wait_asynccnt 0 + barrier makes it visible block-wide.
__device__ __forceinline__
void stage_indices_async(const int* __restrict__ rows, const int* __restrict__ cols,
                         int base, int nE, int tid, int* lrows, int* lcols) {
    if (tid < nE) {
        unsigned l0 = (unsigned)(uintptr_t)&lrows[tid];
        unsigned l1 = (unsigned)(uintptr_t)&lcols[tid];
        const int* g0 = rows + base + tid;
        const int* g1 = cols + base + tid;
        asm volatile("global_load_async_to_lds_b32 %0, %1, off"
                     :: "v"(l0), "v"(g0) : "memory");
        asm volatile("global_load_async_to_lds_b32 %0, %1, off"
                     :: "v"(l1), "v"(g1) : "memory");
    }
    asm volatile("s_wait_asynccnt 0x0" ::: "memory");
    __syncthreads();
}

// Symmetrized bipartite SpMM over one chunk: for edge (u,b):
//   tmp[u] += feat[nA+b],  tmp[nA+b] += feat[u]
// k = tid&63 is fixed per thread -> coalesced 128B per wave32; edge indices
// come from LDS; next iteration's gather rows are prefetched.
__global__ __launch_bounds__(256)
void edge_scatter_sym_k(const int* __restrict__ rows, const int* __restrict__ cols,
                        int E, const float* __restrict__ feat,
                        float* __restrict__ tmp, int nA) {
    __shared__ int lrows[CHUNK];
    __shared__ int lcols[CHUNK];
    int base = blockIdx.x * CHUNK;
    int nE = E - base; if (nE > CHUNK) nE = CHUNK;
    int tid = threadIdx.x;

    stage_indices_async(rows, cols, base, nE, tid, lrows, lcols);

    int k = tid & 63;
    for (int e = tid >> 6; e < nE; e += 4) {
        int u = lrows[e];
        int b = lcols[e] + nA;
        int en = e + 4;
        if (en < nE) {
            __builtin_prefetch(&feat[(size_t)lrows[en] * D64 + k], 0, 3);
            __builtin_prefetch(&feat[((size_t)lcols[en] + nA) * D64 + k], 0, 3);
        }
        float fu = feat[(size_t)u * D64 + k];
        float fb = feat[(size_t)b * D64 + k];
        atomicAdd(&tmp[(size_t)u * D64 + k], fb);
        atomicAdd(&tmp[(size_t)b * D64 + k], fu);
    }
}

// dst[row[e]*stride + k] += src[col[e]*64 + k]   (BI aggregation, unweighted;
// scaled by 1/bundle_size afterwards)
__global__ __launch_bounds__(256)
void edge_gather_add_k(const int* __restrict__ rows, const int* __restrict__ cols,
                       int E, const float* __restrict__ src,
                       float* __restrict__ dst, int dstStride) {
    __shared__ int lrows[CHUNK];
    __shared__ int lcols[CHUNK];
    int base = blockIdx.x * CHUNK;
    int nE = E - base; if (nE > CHUNK) nE = CHUNK;
    int tid = threadIdx.x;

    stage_indices_async(rows, cols, base, nE, tid, lrows, lcols);

    int k = tid & 63;
    for (int e = tid >> 6; e < nE; e += 4) {
        int b = lrows[e];
        int i = lcols[e];
        int en = e + 4;
        if (en < nE)
            __builtin_prefetch(&src[(size_t)lcols[en] * D64 + k], 0, 3);
        atomicAdd(&dst[(size_t)b * dstStride + k], src[(size_t)i * D64 + k]);
    }
}

// acc += tmp_row / max(||tmp_row||2, 1e-12).  One wave32 per row: lane holds
// k and k+32; sum-of-squares reduced with __shfl_xor across the wave.
__global__ __launch_bounds__(256)
void normalize_accum_k(const float* __restrict__ t, int n, int nA,
                       float* __restrict__ accA, int strideA,
                       float* __restrict__ accB, int strideB) {
    int wave = threadIdx.x >> 5;
    int lane = threadIdx.x & 31;
    int row  = blockIdx.x * (blockDim.x >> 5) + wave;
    if (row >= n) return;
    const float* src = t + (size_t)row * D64;
    float x0 = src[lane];
    float x1 = src[lane + 32];
    float s = x0 * x0 + x1 * x1;
    #pragma unroll
    for (int m = 16; m > 0; m >>= 1) s += __shfl_xor(s, m, 32);
    float sc = 1.0f / fmaxf(sqrtf(s), 1e-12f);
    float* dst = (row < nA) ? (accA + (size_t)row * strideA)
                            : (accB + (size_t)(row - nA) * strideB);
    dst[lane]      += x0 * sc;
    dst[lane + 32] += x1 * sc;
}

__global__ __launch_bounds__(256)
void count_rows_k(const int* __restrict__ rows, int E, float* __restrict__ bsize) {
    int e = blockIdx.x * blockDim.x + threadIdx.x;
    if (e < E) atomicAdd(&bsize[rows[e]], 1.0f);
}

static inline int nblk(long n) { return (int)((n + 255) / 256); }

extern "C" void kernel_launch(void* const* d_in, const int* in_sizes, int n_in,
                              void* d_out, int out_size, void* d_ws, size_t ws_size,
                              hipStream_t stream) {
    const float* usersF   = (const float*)d_in[0];
    const float* itemsF   = (const float*)d_in[1];
    const float* bundlesF = (const float*)d_in[2];
    const int* ui_row = (const int*)d_in[3];
    const int* ui_col = (const int*)d_in[4];
    const int* ub_row = (const int*)d_in[5];
    const int* ub_col = (const int*)d_in[6];
    const int* bi_row = (const int*)d_in[7];
    const int* bi_col = (const int*)d_in[8];
    const int E_UI = in_sizes[3];
    const int E_UB = in_sizes[5];
    const int E_BI = in_sizes[7];
    const int num_layers = 2;   // graph capture needs a fixed trip count; setup uses 2

    float* out  = (float*)d_out;
    float* outU = out;                         // [NU,128]: cols 0..63 = IL_u, 64..127 = BL_u
    float* outB = out + (size_t)NU_C * 128;    // [NB,128]: cols 0..63 = IL_b, 64..127 = BL_b

    // workspace layout
    float* feat  = (float*)d_ws;                           // (NU+NI)*64
    float* tmp   = feat + (size_t)(NU_C + NI_C) * D64;     // (NU+NI)*64
    float* accI  = tmp  + (size_t)(NU_C + NI_C) * D64;     // NI*64
    float* bsize = accI + (size_t)NI_C * D64;              // NB
    (void)ws_size; (void)n_in; (void)out_size;

    const int T = 256;

    // ---------------- item-level propagation over the UI graph ----------------
    copy_rows4_k<<<nblk((long)NU_C * 16), T, 0, stream>>>((float4*)feat, 16, (const float4*)usersF, NU_C);
    copy_rows4_k<<<nblk((long)NI_C * 16), T, 0, stream>>>((float4*)(feat + (size_t)NU_C * D64), 16, (const float4*)itemsF, NI_C);
    copy_rows4_k<<<nblk((long)NU_C * 16), T, 0, stream>>>((float4*)outU, 32, (const float4*)usersF, NU_C);   // acc layer-0
    copy_rows4_k<<<nblk((long)NI_C * 16), T, 0, stream>>>((float4*)accI, 16, (const float4*)itemsF, NI_C);

    {
        float* f = feat; float* t = tmp;
        const int n = NU_C + NI_C;
        for (int L = 0; L < num_layers; ++L) {
            fill_zero4_k<<<nblk((long)n * 16), T, 0, stream>>>((float4*)t, (long)n * 16);
            edge_scatter_sym_k<<<(E_UI + CHUNK - 1) / CHUNK, T, 0, stream>>>(ui_row, ui_col, E_UI, f, t, NU_C);
            normalize_accum_k<<<(n + 7) / 8, T, 0, stream>>>(t, n, NU_C, outU, 128, accI, D64);
            float* s = f; f = t; t = s;
        }
    }

    // ---------------- bundle rep from row-normalized BI graph ----------------
    fill_zero4_k<<<nblk(NB_C / 4), T, 0, stream>>>((float4*)bsize, NB_C / 4);
    count_rows_k<<<nblk(E_BI), T, 0, stream>>>(bi_row, E_BI, bsize);
    zero_rows4_k<<<nblk((long)NB_C * 16), T, 0, stream>>>((float4*)outB, 32, NB_C);
    edge_gather_add_k<<<(E_BI + CHUNK - 1) / CHUNK, T, 0, stream>>>(bi_row, bi_col, E_BI, accI, outB, 128);
    scale_rows4_k<<<nblk((long)NB_C * 16), T, 0, stream>>>((float4*)outB, 32, bsize, NB_C);

    // ---------------- bundle-level propagation over the UB graph ----------------
    copy_rows4_k<<<nblk((long)NU_C * 16), T, 0, stream>>>((float4*)feat, 16, (const float4*)usersF, NU_C);
    copy_rows4_k<<<nblk((long)NB_C * 16), T, 0, stream>>>((float4*)(feat + (size_t)NU_C * D64), 16, (const float4*)bundlesF, NB_C);
    copy_rows4_k<<<nblk((long)NU_C * 16), T, 0, stream>>>((float4*)(outU + 64), 32, (const float4*)usersF, NU_C);
    copy_rows4_k<<<nblk((long)NB_C * 16), T, 0, stream>>>((float4*)(outB + 64), 32, (const float4*)bundlesF, NB_C);

    {
        float* f = feat; float* t = tmp;
        const int n = NU_C + NB_C;
        for (int L = 0; L < num_layers; ++L) {
            fill_zero4_k<<<nblk((long)n * 16), T, 0, stream>>>((float4*)t, (long)n * 16);
            edge_scatter_sym_k<<<(E_UB + CHUNK - 1) / CHUNK, T, 0, stream>>>(ub_row, ub_col, E_UB, f, t, NU_C);
            normalize_accum_k<<<(n + 7) / 8, T, 0, stream>>>(t, n, NU_C, outU + 64, 128, outB + 64, 128);
            float* s = f; f = t; t = s;
        }
    }
}